// CDF_26697516712237
// MI455X (gfx1250) — compile-verified
//
#include <hip/hip_runtime.h>
#include <math.h>

// out[i,j] = order[ clamp(floor(Phi(noise[i,j]) * N), 0, N-1), j ]
// noise/out: [BATCH, 256] f32; order: [N_TRAIN, 256] f32 (column-sorted table).
//
// Memory-bound gather kernel tuned for MI455X:
//  - 102.4 MB table fits in the 192 MB L2 -> warm it with global_prefetch_b8
//  - streamed noise/out use non-temporal hints so they don't evict the table
//  - float4 per thread: coalesced b128 stream I/O + 4 independent gathers in flight

typedef float v4f __attribute__((ext_vector_type(4)));

#define NDIM 256

__global__ __launch_bounds__(256) void cdf_gather_kernel(
    const float* __restrict__ noise,
    const float* __restrict__ order,
    float* __restrict__ out,
    long total4,          // number of float4 groups = batch*256/4
    int n_train)
{
    const long nthreads = (long)gridDim.x * blockDim.x;
    const long tid = (long)blockIdx.x * blockDim.x + threadIdx.x;

    // ---- L2 warm-up: each thread prefetches 128B line(s) of the table ----
    // (emits global_prefetch_b8; fire-and-forget, no counter traffic)
    {
        const char* tbl = (const char*)order;
        const size_t tbl_bytes = (size_t)n_train * NDIM * sizeof(float);
        for (size_t off = (size_t)tid * 128; off < tbl_bytes;
             off += (size_t)nthreads * 128) {
            __builtin_prefetch(tbl + off, 0, 3);
        }
    }

    const float nf  = (float)n_train;
    const int   nm1 = n_train - 1;
    const float inv_sqrt2 = 0.70710678118654752440f;

    for (long k4 = tid; k4 < total4; k4 += nthreads) {
        // 4 consecutive columns of one row: coalesced 16B load (NT hint)
        v4f x = __builtin_nontemporal_load((const v4f*)noise + k4);

        const long k   = k4 << 2;            // flat element index
        const int  col = (int)(k & (NDIM - 1));

        v4f r;
#pragma unroll
        for (int e = 0; e < 4; ++e) {
            // ndtr(x) = 0.5 * erfc(-x / sqrt(2))  (matches jax.scipy ndtr)
            float u = 0.5f * erfcf(-x[e] * inv_sqrt2);
            int idx = (int)floorf(u * nf);
            idx = idx < 0 ? 0 : (idx > nm1 ? nm1 : idx);
            // per-column gather: random row, fixed column -> L2-resident hit
            r[e] = order[(size_t)idx * NDIM + (size_t)(col + e)];
        }

        // streamed write-once output: NT store, don't pollute L2
        __builtin_nontemporal_store(r, (v4f*)out + k4);
    }
}

extern "C" void kernel_launch(void* const* d_in, const int* in_sizes, int n_in,
                              void* d_out, int out_size, void* d_ws, size_t ws_size,
                              hipStream_t stream) {
    const float* noise = (const float*)d_in[0];  // [batch, 256]
    const float* order = (const float*)d_in[1];  // [n_train, 256]
    float* out = (float*)d_out;

    const int batch   = in_sizes[0] / NDIM;
    const int n_train = in_sizes[1] / NDIM;

    const long total4 = ((long)batch * NDIM) >> 2;   // 1,048,576 for ref sizes

    const int block = 256;                           // 8 wave32 per block
    long grid_l = (total4 + block - 1) / block;      // one float4 per thread
    if (grid_l > 8192) grid_l = 8192;                // grid-stride beyond this
    const int grid = (int)(grid_l > 0 ? grid_l : 1);

    cdf_gather_kernel<<<grid, block, 0, stream>>>(noise, order, out,
                                                  total4, n_train);
}